// Encoder_26474178412960
// MI455X (gfx1250) — compile-verified
//
#include <hip/hip_runtime.h>

#define N_NODES 50000
#define N_EDGES 800000
#define DIM 128
#define N_LAYERS 3
#define N_GRAPHS 256
#define BN_EPS 1e-5f

typedef __attribute__((ext_vector_type(2))) float v2f;
typedef __attribute__((ext_vector_type(8))) float v8f;
typedef __attribute__((ext_vector_type(4))) int   v4i;

#define LDS_STRIDE 132  // 128 + 4 floats pad: column reads hit distinct banks

#if __has_builtin(__builtin_amdgcn_global_load_async_to_lds_b128)
#define HAVE_ASYNC_LDS 1
typedef __attribute__((address_space(1))) v4i* as1_v4i;
typedef __attribute__((address_space(3))) v4i* as3_v4i;
#else
#define HAVE_ASYNC_LDS 0
#endif

static __device__ __forceinline__ void wait_asynccnt0()
{
#if HAVE_ASYNC_LDS
#if __has_builtin(__builtin_amdgcn_s_wait_asynccnt)
    __builtin_amdgcn_s_wait_asynccnt(0);
#else
    asm volatile("s_wait_asynccnt 0" ::: "memory");
#endif
#endif
}

// ---------------------------------------------------------------------------
// Fused GEMM: out[M,128] = epilogue( (A [+ Aadd]) @ W )
//   MODE 0: plain                                (h = x @ fc_w)
//   MODE 1: relu((h+agg) @ w1 + b1)              (MLP layer 1)
//   MODE 2: relu(t @ w2 + b2)  + column sum/sumsq atomics for BN
// W is [128,128] row-major (k, n). Uses V_WMMA_F32_16X16X4_F32.
// A tile staged to LDS via GLOBAL_LOAD_ASYNC_TO_LDS_B128 (modes 0/2).
// ---------------------------------------------------------------------------
template<int MODE>
__global__ __launch_bounds__(256)
void gemm_kernel(const float* __restrict__ A,
                 const float* __restrict__ Aadd,
                 const float* __restrict__ W,
                 const float* __restrict__ bias,
                 float* __restrict__ out,
                 float* __restrict__ stats,
                 int M)
{
    __shared__ float Alds[32 * LDS_STRIDE];
    const int tid = threadIdx.x;
    const int row_base = blockIdx.x * 32;

    // Warm the WGP$ with the (shared, 64KB) weight matrix: 2 cachelines/thread.
    __builtin_prefetch(W + tid * 32, 0, 3);
    __builtin_prefetch(W + (256 + tid) * 32, 0, 3);

#if HAVE_ASYNC_LDS
    if (MODE != 1) {
        // Async global->LDS staging: bypasses VGPRs, tracked by ASYNCcnt.
        // Tail rows are clamped (not zeroed): padded-row results are masked
        // at the epilogue, and clamped data is finite, so this is safe.
        #pragma unroll
        for (int i = 0; i < 4; ++i) {
            int idx = tid + i * 256;            // float4 index, 1024 total
            int r  = idx >> 5;                  // 32 float4 per row
            int c4 = idx & 31;
            int rg = row_base + r;
            if (rg >= M) rg = M - 1;
            const float* gp = A + (size_t)rg * DIM + c4 * 4;
            float* lp = &Alds[r * LDS_STRIDE + c4 * 4];
            __builtin_amdgcn_global_load_async_to_lds_b128((as1_v4i)gp,
                                                           (as3_v4i)lp, 0, 0);
        }
        wait_asynccnt0();
    } else
#endif
    {
        // Sync staging (fuses the h+agg add for MODE 1), zero-padded tail.
        #pragma unroll
        for (int i = 0; i < 4; ++i) {
            int idx = tid + i * 256;            // float4 index, 1024 total
            int r  = idx >> 5;
            int c4 = idx & 31;
            int rg = row_base + r;
            float4 v = make_float4(0.f, 0.f, 0.f, 0.f);
            if (rg < M) {
                v = ((const float4*)(A + (size_t)rg * DIM))[c4];
                if (MODE == 1) {
                    float4 w = ((const float4*)(Aadd + (size_t)rg * DIM))[c4];
                    v.x += w.x; v.y += w.y; v.z += w.z; v.w += w.w;
                }
            }
            float* p = &Alds[r * LDS_STRIDE + c4 * 4];
            p[0] = v.x; p[1] = v.y; p[2] = v.z; p[3] = v.w;
        }
    }
    __syncthreads();

    const int wave = tid >> 5;              // 0..7 -> N strip
    const int lane = tid & 31;
    const int half = lane >> 4;             // K sub-select per ISA A/B layout
    const int l16  = lane & 15;
    const int n0   = wave * 16;

    v8f acc0 = {};                          // rows row_base+0..15
    v8f acc1 = {};                          // rows row_base+16..31
    const float* Ap0 = &Alds[l16 * LDS_STRIDE];
    const float* Ap1 = &Alds[(16 + l16) * LDS_STRIDE];
    const float* Wp  = W + n0 + l16;

    #pragma unroll
    for (int k = 0; k < DIM; k += 4) {
        const int kk = k + 2 * half;
        v2f a0, a1, b;
        // A 16x4 f32 frag: lanes 0-15 carry K=k..k+1, lanes 16-31 K=k+2..k+3
        a0.x = Ap0[kk];        a0.y = Ap0[kk + 1];
        a1.x = Ap1[kk];        a1.y = Ap1[kk + 1];
        // B 4x16 f32 frag, mirrored layout over K
        b.x  = Wp[kk * DIM];   b.y  = Wp[(kk + 1) * DIM];
        acc0 = __builtin_amdgcn_wmma_f32_16x16x4_f32(false, a0, false, b,
                                                     (short)0, acc0, false, false);
        acc1 = __builtin_amdgcn_wmma_f32_16x16x4_f32(false, a1, false, b,
                                                     (short)0, acc1, false, false);
    }

    const int col = n0 + l16;
    const float bv = (MODE != 0) ? bias[col] : 0.f;
    float s = 0.f, s2 = 0.f;
    #pragma unroll
    for (int r = 0; r < 8; ++r) {
        // C/D layout: VGPR r = row r (lanes 0-15) / row r+8 (lanes 16-31)
        int row0 = row_base + r + 8 * half;
        float v = acc0[r];
        if (MODE != 0) { v += bv; v = v > 0.f ? v : 0.f; }
        if (row0 < M) {
            out[(size_t)row0 * DIM + col] = v;
            s += v; s2 += v * v;
        }
        int row1 = row_base + 16 + r + 8 * half;
        float w = acc1[r];
        if (MODE != 0) { w += bv; w = w > 0.f ? w : 0.f; }
        if (row1 < M) {
            out[(size_t)row1 * DIM + col] = w;
            s += w; s2 += w * w;
        }
    }
    if (MODE == 2) {
        __hip_atomic_fetch_add(&stats[col], s,
                               __ATOMIC_RELAXED, __HIP_MEMORY_SCOPE_AGENT);
        __hip_atomic_fetch_add(&stats[DIM + col], s2,
                               __ATOMIC_RELAXED, __HIP_MEMORY_SCOPE_AGENT);
    }
}

// ---------------------------------------------------------------------------
__global__ void zero4_kernel(float4* __restrict__ p, int n4)
{
    int i = blockIdx.x * blockDim.x + threadIdx.x;
    int stride = gridDim.x * blockDim.x;
    for (; i < n4; i += stride) p[i] = make_float4(0.f, 0.f, 0.f, 0.f);
}

// One wave per edge: lane grabs a float4 of h[src], 4 f32 atomics into agg[dst].
__global__ __launch_bounds__(256)
void scatter_add_kernel(const float* __restrict__ h,
                        const int* __restrict__ src,
                        const int* __restrict__ dst,
                        float* __restrict__ agg, int n_edges)
{
    int e    = (blockIdx.x * blockDim.x + threadIdx.x) >> 5;
    int lane = threadIdx.x & 31;
    if (e >= n_edges) return;
    int si = src[e], di = dst[e];
    float4 v = ((const float4*)(h + (size_t)si * DIM))[lane];
    float* p = agg + (size_t)di * DIM + lane * 4;
    __hip_atomic_fetch_add(p + 0, v.x, __ATOMIC_RELAXED, __HIP_MEMORY_SCOPE_AGENT);
    __hip_atomic_fetch_add(p + 1, v.y, __ATOMIC_RELAXED, __HIP_MEMORY_SCOPE_AGENT);
    __hip_atomic_fetch_add(p + 2, v.z, __ATOMIC_RELAXED, __HIP_MEMORY_SCOPE_AGENT);
    __hip_atomic_fetch_add(p + 3, v.w, __ATOMIC_RELAXED, __HIP_MEMORY_SCOPE_AGENT);
}

// Fold BN (biased variance, batch stats) into per-column scale/shift.
__global__ void bn_finalize_kernel(const float* __restrict__ stats,
                                   const float* __restrict__ gamma,
                                   const float* __restrict__ beta,
                                   float* __restrict__ scaleshift)
{
    int j = threadIdx.x;                    // 128 threads
    const float invN = 1.0f / (float)N_NODES;
    float mean = stats[j] * invN;
    float var  = stats[DIM + j] * invN - mean * mean;
    var = var > 0.f ? var : 0.f;
    float sc = gamma[j] * rsqrtf(var + BN_EPS);
    scaleshift[j]       = sc;
    scaleshift[DIM + j] = beta[j] - mean * sc;
}

// h = z*scale+shift, also written to its slice of the concat output.
__global__ __launch_bounds__(256)
void bn_apply_kernel(const float* __restrict__ z,
                     const float* __restrict__ scaleshift,
                     float* __restrict__ h,
                     float* __restrict__ out_concat,
                     int layer)
{
    long n4 = (long)N_NODES * DIM / 4;
    long i  = (long)blockIdx.x * blockDim.x + threadIdx.x;
    long stride = (long)gridDim.x * blockDim.x;
    for (; i < n4; i += stride) {
        int node = (int)(i >> 5);
        int c4   = (int)(i & 31);
        float4 v  = ((const float4*)z)[i];
        float4 sc = ((const float4*)scaleshift)[c4];
        float4 sh = ((const float4*)(scaleshift + DIM))[c4];
        float4 r;
        r.x = v.x * sc.x + sh.x;
        r.y = v.y * sc.y + sh.y;
        r.z = v.z * sc.z + sh.z;
        r.w = v.w * sc.w + sh.w;
        ((float4*)h)[i] = r;
        ((float4*)(out_concat + (size_t)node * (N_LAYERS * DIM) + layer * DIM))[c4] = r;
    }
}

// global_add_pool: wave per node, atomics into pooled[batch[node]].
__global__ __launch_bounds__(256)
void pool_kernel(const float* __restrict__ h,
                 const int* __restrict__ batch,
                 float* __restrict__ pooled)
{
    int node = (blockIdx.x * blockDim.x + threadIdx.x) >> 5;
    int lane = threadIdx.x & 31;
    if (node >= N_NODES) return;
    int b = batch[node];
    float4 v = ((const float4*)(h + (size_t)node * DIM))[lane];
    float* p = pooled + (size_t)b * DIM + lane * 4;
    __hip_atomic_fetch_add(p + 0, v.x, __ATOMIC_RELAXED, __HIP_MEMORY_SCOPE_AGENT);
    __hip_atomic_fetch_add(p + 1, v.y, __ATOMIC_RELAXED, __HIP_MEMORY_SCOPE_AGENT);
    __hip_atomic_fetch_add(p + 2, v.z, __ATOMIC_RELAXED, __HIP_MEMORY_SCOPE_AGENT);
    __hip_atomic_fetch_add(p + 3, v.w, __ATOMIC_RELAXED, __HIP_MEMORY_SCOPE_AGENT);
}

// ---------------------------------------------------------------------------
extern "C" void kernel_launch(void* const* d_in, const int* in_sizes, int n_in,
                              void* d_out, int out_size, void* d_ws, size_t ws_size,
                              hipStream_t stream)
{
    const float* x     = (const float*)d_in[0];
    const int*   eidx  = (const int*)d_in[1];   // [2, E]
    const int*   batch = (const int*)d_in[2];
    const float* fc_w  = (const float*)d_in[3];
    const float* w1    = (const float*)d_in[4];
    const float* b1    = (const float*)d_in[5];
    const float* w2    = (const float*)d_in[6];
    const float* b2    = (const float*)d_in[7];
    const float* gamma = (const float*)d_in[8];
    const float* beta  = (const float*)d_in[9];
    float* out = (float*)d_out;

    const int* src = eidx;
    const int* dst = eidx + N_EDGES;

    const size_t NF = (size_t)N_NODES * DIM;
    float* h          = (float*)d_ws;          // [N, 128]
    float* agg        = h + NF;                // [N, 128]; reused as z after GEMM1
    float* t          = agg + NF;              // [N, 128]
    float* stats      = t + NF;                // [256] sum | sumsq
    float* scaleshift = stats + 2 * DIM;       // [256] scale | shift

    float* pooled = out;                       // [256, 128]
    float* concat = out + (size_t)N_GRAPHS * DIM;  // [N, 384]

    const int gemm_grid    = (N_NODES + 31) / 32;               // 1563
    const int scatter_grid = (N_EDGES * 32 + 255) / 256;        // 100000
    const int nf4          = (int)(NF / 4);                     // 1.6M

    // h = x @ fc_w
    gemm_kernel<0><<<gemm_grid, 256, 0, stream>>>(x, nullptr, fc_w, nullptr,
                                                  h, nullptr, N_NODES);

    for (int l = 0; l < N_LAYERS; ++l) {
        // agg = segment_sum(h[src], dst)
        zero4_kernel<<<2048, 256, 0, stream>>>((float4*)agg, nf4);
        scatter_add_kernel<<<scatter_grid, 256, 0, stream>>>(h, src, dst, agg, N_EDGES);

        // t = relu((h + agg) @ w1[l] + b1[l])
        gemm_kernel<1><<<gemm_grid, 256, 0, stream>>>(h, agg,
                                                      w1 + (size_t)l * DIM * DIM,
                                                      b1 + (size_t)l * DIM,
                                                      t, nullptr, N_NODES);

        // z = relu(t @ w2[l] + b2[l])  (z aliases agg) + BN column stats
        zero4_kernel<<<1, 64, 0, stream>>>((float4*)stats, 2 * DIM / 4);
        gemm_kernel<2><<<gemm_grid, 256, 0, stream>>>(t, nullptr,
                                                      w2 + (size_t)l * DIM * DIM,
                                                      b2 + (size_t)l * DIM,
                                                      agg, stats, N_NODES);

        // h = gamma*(z-mean)*rsqrt(var+eps)+beta; also write concat slice
        bn_finalize_kernel<<<1, DIM, 0, stream>>>(stats,
                                                  gamma + (size_t)l * DIM,
                                                  beta + (size_t)l * DIM,
                                                  scaleshift);
        bn_apply_kernel<<<6250, 256, 0, stream>>>(agg, scaleshift, h, concat, l);
    }

    // pooled = segment_sum(h_last, batch)
    zero4_kernel<<<32, 256, 0, stream>>>((float4*)pooled, N_GRAPHS * DIM / 4);
    pool_kernel<<<(N_NODES * 32 + 255) / 256, 256, 0, stream>>>(h, batch, pooled);
}